// PairMixing_14516989461264
// MI455X (gfx1250) — compile-verified
//
#include <hip/hip_runtime.h>

// ---------------------------------------------------------------------------
// Compile-time Clebsch-Gordan (real spherical harmonics) tables
// ---------------------------------------------------------------------------
typedef __attribute__((ext_vector_type(2))) float v2f;
typedef __attribute__((ext_vector_type(8))) float v8f;

constexpr double FACT[13] = {1.0, 1.0, 2.0, 6.0, 24.0, 120.0, 720.0, 5040.0,
                             40320.0, 362880.0, 3628800.0, 39916800.0, 479001600.0};

constexpr double csqrt(double x) {
    if (x <= 0.0) return 0.0;
    double g = x > 1.0 ? x : 1.0;
    for (int i = 0; i < 60; ++i) g = 0.5 * (g + x / g);
    return g;
}

constexpr double clebsch(int j1, int m1, int j2, int m2, int j3, int m3) {
    if (m1 + m2 != m3) return 0.0;
    int dj = j1 > j2 ? j1 - j2 : j2 - j1;
    if (j3 < dj || j3 > j1 + j2) return 0.0;
    double pref = csqrt((2.0 * j3 + 1.0) * FACT[j3 + j1 - j2] * FACT[j3 - j1 + j2] *
                        FACT[j1 + j2 - j3] / FACT[j1 + j2 + j3 + 1]);
    pref *= csqrt(FACT[j3 + m3] * FACT[j3 - m3] * FACT[j1 - m1] * FACT[j1 + m1] *
                  FACT[j2 - m2] * FACT[j2 + m2]);
    int kmin = 0;
    if (j2 - j3 - m1 > kmin) kmin = j2 - j3 - m1;
    if (j1 - j3 + m2 > kmin) kmin = j1 - j3 + m2;
    int kmax = j1 + j2 - j3;
    if (j1 - m1 < kmax) kmax = j1 - m1;
    if (j2 + m2 < kmax) kmax = j2 + m2;
    double s = 0.0;
    for (int k = kmin; k <= kmax; ++k) {
        double term = 1.0 / (FACT[k] * FACT[j1 + j2 - j3 - k] * FACT[j1 - m1 - k] *
                             FACT[j2 + m2 - k] * FACT[j3 - j2 + m1 + k] * FACT[j3 - j1 - m2 + k]);
        s += (k & 1) ? -term : term;
    }
    return pref * s;
}

struct Uent { int col; double re; double im; };

// rows of unitary complex->real SH transform; each row has <= 2 nonzeros
constexpr int urow(int l, int a, Uent* o) {
    const double is2 = 0.70710678118654752440;
    int rm = a - l;
    if (rm == 0) { o[0].col = l; o[0].re = 1.0; o[0].im = 0.0; return 1; }
    int m = rm > 0 ? rm : -rm;
    double sgn = (m & 1) ? -1.0 : 1.0;
    if (rm > 0) {
        o[0].col = l + m; o[0].re = sgn * is2; o[0].im = 0.0;
        o[1].col = l - m; o[1].re = is2;       o[1].im = 0.0;
    } else {
        o[0].col = l + m; o[0].re = 0.0; o[0].im = -sgn * is2;
        o[1].col = l - m; o[1].re = 0.0; o[1].im = is2;
    }
    return 2;
}

struct DenseCG { double v[441]; };  // max (2*3+1)*(2*3+1)*(2*4+1)

constexpr DenseCG make_cg(int l1, int l2, int l3) {
    int d1 = 2 * l1 + 1, d2 = 2 * l2 + 1, d3 = 2 * l3 + 1;
    double cgd[441] = {};
    for (int i = 0; i < d1; ++i)
        for (int j = 0; j < d2; ++j) {
            int m1 = i - l1, m2 = j - l2, m3 = m1 + m2;
            if (m3 >= -l3 && m3 <= l3)
                cgd[(i * d2 + j) * d3 + (m3 + l3)] = clebsch(l1, m1, l2, m2, l3, m3);
        }
    double vr[441] = {}, vi[441] = {};
    double maxr = 0.0, maxi = 0.0;
    for (int a = 0; a < d1; ++a) {
        Uent ua[2] = {}; int na = urow(l1, a, ua);
        for (int b = 0; b < d2; ++b) {
            Uent ub[2] = {}; int nb = urow(l2, b, ub);
            for (int c = 0; c < d3; ++c) {
                Uent uc[2] = {}; int nc = urow(l3, c, uc);
                double re = 0.0, im = 0.0;
                for (int ia = 0; ia < na; ++ia)
                    for (int ib = 0; ib < nb; ++ib) {
                        double pr = ua[ia].re * ub[ib].re - ua[ia].im * ub[ib].im;
                        double pi = ua[ia].re * ub[ib].im + ua[ia].im * ub[ib].re;
                        for (int ic = 0; ic < nc; ++ic) {
                            double cg = cgd[(ua[ia].col * d2 + ub[ib].col) * d3 + uc[ic].col];
                            if (cg != 0.0) {
                                // * conj(uc)
                                re += cg * (pr * uc[ic].re + pi * uc[ic].im);
                                im += cg * (pi * uc[ic].re - pr * uc[ic].im);
                            }
                        }
                    }
                int idx = (a * d2 + b) * d3 + c;
                vr[idx] = re; vi[idx] = im;
                double ar = re < 0 ? -re : re, ai = im < 0 ? -im : im;
                if (ar > maxr) maxr = ar;
                if (ai > maxi) maxi = ai;
            }
        }
    }
    DenseCG outp = {};
    bool useR = (maxr >= maxi);
    for (int k = 0; k < d1 * d2 * d3; ++k) outp.v[k] = useR ? vr[k] : vi[k];
    return outp;
}

// couplings in reference order: for l1: for l2: for lout in [|l1-l2|, min(l1+l2,4)]
constexpr DenseCG CG0  = make_cg(0,0,0);
constexpr DenseCG CG1  = make_cg(0,1,1);
constexpr DenseCG CG2  = make_cg(0,2,2);
constexpr DenseCG CG3  = make_cg(0,3,3);
constexpr DenseCG CG4  = make_cg(1,0,1);
constexpr DenseCG CG5  = make_cg(1,1,0);
constexpr DenseCG CG6  = make_cg(1,1,1);
constexpr DenseCG CG7  = make_cg(1,1,2);
constexpr DenseCG CG8  = make_cg(1,2,1);
constexpr DenseCG CG9  = make_cg(1,2,2);
constexpr DenseCG CG10 = make_cg(1,2,3);
constexpr DenseCG CG11 = make_cg(1,3,2);
constexpr DenseCG CG12 = make_cg(1,3,3);
constexpr DenseCG CG13 = make_cg(1,3,4);
constexpr DenseCG CG14 = make_cg(2,0,2);
constexpr DenseCG CG15 = make_cg(2,1,1);
constexpr DenseCG CG16 = make_cg(2,1,2);
constexpr DenseCG CG17 = make_cg(2,1,3);
constexpr DenseCG CG18 = make_cg(2,2,0);
constexpr DenseCG CG19 = make_cg(2,2,1);
constexpr DenseCG CG20 = make_cg(2,2,2);
constexpr DenseCG CG21 = make_cg(2,2,3);
constexpr DenseCG CG22 = make_cg(2,2,4);
constexpr DenseCG CG23 = make_cg(2,3,1);
constexpr DenseCG CG24 = make_cg(2,3,2);
constexpr DenseCG CG25 = make_cg(2,3,3);
constexpr DenseCG CG26 = make_cg(2,3,4);
constexpr DenseCG CG27 = make_cg(3,0,3);
constexpr DenseCG CG28 = make_cg(3,1,2);
constexpr DenseCG CG29 = make_cg(3,1,3);
constexpr DenseCG CG30 = make_cg(3,1,4);
constexpr DenseCG CG31 = make_cg(3,2,1);
constexpr DenseCG CG32 = make_cg(3,2,2);
constexpr DenseCG CG33 = make_cg(3,2,3);
constexpr DenseCG CG34 = make_cg(3,2,4);
constexpr DenseCG CG35 = make_cg(3,3,0);
constexpr DenseCG CG36 = make_cg(3,3,1);
constexpr DenseCG CG37 = make_cg(3,3,2);
constexpr DenseCG CG38 = make_cg(3,3,3);
constexpr DenseCG CG39 = make_cg(3,3,4);

constexpr int NCP = 40;
constexpr int CAP = 5200;  // >= dense total (5110), so no overflow possible
constexpr int L1A[NCP]   = {0,0,0,0, 1,1,1,1,1,1,1,1,1,1, 2,2,2,2,2,2,2,2,2,2,2,2,2, 3,3,3,3,3,3,3,3,3,3,3,3,3};
constexpr int L2A[NCP]   = {0,1,2,3, 0,1,1,1,2,2,2,3,3,3, 0,1,1,1,2,2,2,2,2,3,3,3,3, 0,1,1,1,2,2,2,2,3,3,3,3,3};
constexpr int LOA[NCP]   = {0,1,2,3, 1,0,1,2,1,2,3,2,3,4, 2,1,2,3,0,1,2,3,4,1,2,3,4, 3,2,3,4,1,2,3,4,0,1,2,3,4};
constexpr int XBASE[4]   = {0, 1, 4, 9};        // m-slot (32ch) offsets of l=0..3 in x/y
constexpr int OBASE[5]   = {0, 1, 4, 9, 16};    // m-slot offsets of l=0..4 in out

struct CGTable {
    int n;
    int cs[NCP + 1];
    int lout[NCP];
    int obase[NCP];
    unsigned char xi[CAP], yi[CAP], mi[CAP];
    float cf[CAP];
};

constexpr CGTable build_table() {
    CGTable t = {};
    const DenseCG* cgs[NCP] = {
        &CG0,&CG1,&CG2,&CG3,&CG4,&CG5,&CG6,&CG7,&CG8,&CG9,
        &CG10,&CG11,&CG12,&CG13,&CG14,&CG15,&CG16,&CG17,&CG18,&CG19,
        &CG20,&CG21,&CG22,&CG23,&CG24,&CG25,&CG26,&CG27,&CG28,&CG29,
        &CG30,&CG31,&CG32,&CG33,&CG34,&CG35,&CG36,&CG37,&CG38,&CG39};
    int n = 0;
    for (int i = 0; i < NCP; ++i) {
        t.cs[i] = n;
        int l1 = L1A[i], l2 = L2A[i], lo = LOA[i];
        t.lout[i] = lo; t.obase[i] = OBASE[lo];
        int d1 = 2 * l1 + 1, d2 = 2 * l2 + 1, d3 = 2 * lo + 1;
        for (int a = 0; a < d1; ++a)
            for (int b = 0; b < d2; ++b)
                for (int c = 0; c < d3; ++c) {
                    double v = cgs[i]->v[(a * d2 + b) * d3 + c];
                    double av = v < 0 ? -v : v;
                    if (av > 1e-12) {
                        t.xi[n] = (unsigned char)(XBASE[l1] + a);
                        t.yi[n] = (unsigned char)(XBASE[l2] + b);
                        t.mi[n] = (unsigned char)c;
                        t.cf[n] = (float)v;
                        ++n;
                    }
                }
    }
    t.cs[NCP] = n; t.n = n;
    return t;
}

constexpr CGTable TAB = build_table();

// ---------------------------------------------------------------------------
// Fully-unrolled per-coupling CG step: all indices/coeffs are literals.
// ---------------------------------------------------------------------------
template <int I>
__device__ __forceinline__ void coupling_step(const float* xv, const float* yv,
                                              const float* fls, float* ov) {
    constexpr int b  = TAB.cs[I];
    constexpr int e_ = TAB.cs[I + 1];
    constexpr int lo = TAB.lout[I];
    constexpr int ob = TAB.obase[I];
    float z[2 * lo + 1];
#pragma unroll
    for (int m = 0; m < 2 * lo + 1; ++m) z[m] = 0.0f;
#pragma unroll
    for (int t = b; t < e_; ++t)
        z[TAB.mi[t]] += TAB.cf[t] * xv[TAB.xi[t]] * yv[TAB.yi[t]];
    float f = fls[I * 32];  // radial filter for this coupling (LDS)
#pragma unroll
    for (int m = 0; m < 2 * lo + 1; ++m) ov[ob + m] += f * z[m];
}

template <int I>
struct RunC {
    static __device__ __forceinline__ void go(const float* xv, const float* yv,
                                              const float* fls, float* ov) {
        coupling_step<I>(xv, yv, fls, ov);
        RunC<I + 1>::go(xv, yv, fls, ov);
    }
};
template <>
struct RunC<NCP> {
    static __device__ __forceinline__ void go(const float*, const float*, const float*, float*) {}
};

// ---------------------------------------------------------------------------
// Kernel 1: repack W [40,64,32] into per-WMMA B fragments (float2 per lane).
// B is K=64 x N=1280 with B[f][i*32+c] = W[i,f,c].
// Fragment layout (v_wmma_f32_16x16x4_f32 B, 4x16 f32): lanes 0-15 hold
// K={0,1} in the two VGPRs, lanes 16-31 hold K={2,3}; N = lane%16.
// ---------------------------------------------------------------------------
__global__ void prep_B_kernel(const float* __restrict__ W, float2* __restrict__ Bw) {
    int idx = blockIdx.x * 256 + threadIdx.x;
    if (idx >= 80 * 16 * 32) return;            // 80 n-tiles * 16 k-steps * 32 lanes
    int t  = idx & 31;
    int s  = (idx >> 5) & 15;
    int nt = idx >> 9;
    int col = nt * 16 + (t & 15);               // N column 0..1279
    int i = col >> 5, c = col & 31;             // coupling, channel
    int k0 = 4 * s + ((t >> 4) << 1);           // K base for this lane
    float b0 = W[i * 2048 + k0 * 32 + c];
    float b1 = W[i * 2048 + (k0 + 1) * 32 + c];
    Bw[idx] = make_float2(b0, b1);
}

// ---------------------------------------------------------------------------
// Kernel 2: fused radial GEMM (WMMA f32 16x16x4) + CG tensor product.
// Block = 512 threads (16 waves, wave32), edge tile = 16 (E = 6250*16 exact).
// ---------------------------------------------------------------------------
#define RSTRIDE 68   // padded LDS stride for r tile (bank-conflict free)
#define FSTRIDE 1280

__global__ __launch_bounds__(512) void pairmix_kernel(
    const float* __restrict__ x, const float* __restrict__ y,
    const float* __restrict__ r, const float2* __restrict__ Bw,
    float* __restrict__ out) {
    extern __shared__ float lds[];
    float* rlds = lds;                      // [16][RSTRIDE]
    float* Flds = lds + 16 * RSTRIDE;       // [16][FSTRIDE]

    const int tid  = threadIdx.x;
    const int tile = blockIdx.x;

    // ---- load r tile [16 edges x 64 feats] into LDS (coalesced) ----
    {
        int i0 = tid;            // 1024 floats, 512 threads -> 2 each
        int i1 = tid + 512;
        rlds[(i0 >> 6) * RSTRIDE + (i0 & 63)] = r[tile * 1024 + i0];
        rlds[(i1 >> 6) * RSTRIDE + (i1 & 63)] = r[tile * 1024 + i1];
    }
    __syncthreads();

    // ---- phase 1: F[16 x 1280] = r_tile @ B via v_wmma_f32_16x16x4_f32 ----
    {
        const int wave = tid >> 5;
        const int lane = tid & 31;
        const int m    = lane & 15;
        const int kb   = (lane >> 4) << 1;  // 0 or 2

        v2f afr[16];
#pragma unroll
        for (int s = 0; s < 16; ++s) {      // A fragments: 16 K-steps of 4
            afr[s].x = rlds[m * RSTRIDE + 4 * s + kb];
            afr[s].y = rlds[m * RSTRIDE + 4 * s + kb + 1];
        }
#pragma unroll
        for (int u = 0; u < 5; ++u) {       // 80 n-tiles / 16 waves = 5
            int nt = wave * 5 + u;
            v8f acc = {};
            const float2* bp = Bw + nt * 512 + lane;
#pragma unroll
            for (int s = 0; s < 16; ++s) {
                float2 bv = bp[s * 32];
                v2f b; b.x = bv.x; b.y = bv.y;
                acc = __builtin_amdgcn_wmma_f32_16x16x4_f32(
                    false, afr[s], false, b, (short)0, acc, false, false);
            }
            int col   = nt * 16 + m;
            int mbase = (lane >> 4) << 3;   // rows 0-7 or 8-15
#pragma unroll
            for (int j = 0; j < 8; ++j)
                Flds[(mbase + j) * FSTRIDE + col] = acc[j];
        }
    }
    __syncthreads();

    // ---- phase 2: CG tensor product, one thread per (edge, channel) ----
    {
        const int el = tid >> 5;            // local edge 0..15
        const int c  = tid & 31;            // channel
        const long e = (long)tile * 16 + el;

        const float* xp = x + e * 512 + c;
        const float* yp = y + e * 512 + c;
        float xv[16], yv[16];
#pragma unroll
        for (int j = 0; j < 16; ++j) { xv[j] = xp[j * 32]; yv[j] = yp[j * 32]; }

        float ov[25];
#pragma unroll
        for (int j = 0; j < 25; ++j) ov[j] = 0.0f;

        const float* fls = Flds + el * FSTRIDE + c;
        RunC<0>::go(xv, yv, fls, ov);

        float* op = out + e * 800 + c;
#pragma unroll
        for (int j = 0; j < 25; ++j) op[j * 32] = ov[j];
    }
}

// ---------------------------------------------------------------------------
extern "C" void kernel_launch(void* const* d_in, const int* in_sizes, int n_in,
                              void* d_out, int out_size, void* d_ws, size_t ws_size,
                              hipStream_t stream) {
    (void)in_sizes; (void)n_in; (void)out_size; (void)ws_size;
    const float* x = (const float*)d_in[0];   // [100000, 512]
    const float* y = (const float*)d_in[1];   // [100000, 512]
    const float* r = (const float*)d_in[2];   // [100000, 64]
    const float* W = (const float*)d_in[3];   // [40, 64, 32]
    float* out = (float*)d_out;               // [100000, 800]
    float2* Bw = (float2*)d_ws;               // 80*16*32 float2 = 327,680 B

    prep_B_kernel<<<dim3(160), dim3(256), 0, stream>>>(W, Bw);

    size_t shmem = (size_t)(16 * RSTRIDE + 16 * FSTRIDE) * sizeof(float);  // 86,272 B
    pairmix_kernel<<<dim3(6250), dim3(512), shmem, stream>>>(x, y, r, Bw, out);
}